// PV_Joint_11725260718247
// MI455X (gfx1250) — compile-verified
//
#include <hip/hip_runtime.h>
#include <math.h>

// ---------------- problem constants (match reference) ----------------
#define NB    32          // batch
#define ORIG  256
#define UPD   512
#define NPIX  (UPD*UPD)   // 262144
#define NM    4           // masks
#define RAD   32          // crop half-size
#define CROP  64          // 2*RAD
#define DOT2  100         // dot_size^2 = 10^2
#define COEF  1.5f

typedef float v2f __attribute__((ext_vector_type(2)));
typedef float v8f __attribute__((ext_vector_type(8)));
typedef unsigned int v4u __attribute__((ext_vector_type(4)));
typedef int v4i __attribute__((ext_vector_type(4)));
typedef int v8i __attribute__((ext_vector_type(8)));

// ---------------- wave32 sum via V_WMMA_F32_16X16X4_F32 ----------------
// A: lane L puts partial in K-slot0 (lanes 0-15 -> K=0, lanes 16-31 -> K=2),
// B = ones => D[m,n] = p[m] + p[m+16]. Per-lane sum of the 8 D VGPRs covers
// M=0..7 (lanes<16) / M=8..15 (lanes>=16); one xor-16 shuffle completes it.
__device__ __forceinline__ float wave_sum32(float p) {
#if defined(__gfx1250__) && __has_builtin(__builtin_amdgcn_wmma_f32_16x16x4_f32)
  v2f a;    a.x = p;    a.y = 0.0f;
  v2f ones; ones.x = 1.0f; ones.y = 1.0f;
  v8f c = {0.f,0.f,0.f,0.f,0.f,0.f,0.f,0.f};
  v8f d = __builtin_amdgcn_wmma_f32_16x16x4_f32(false, a, false, ones,
                                                (short)0, c, false, false);
  float t = ((d[0]+d[1])+(d[2]+d[3])) + ((d[4]+d[5])+(d[6]+d[7]));
  t += __shfl_xor(t, 16, 32);
  return t;
#else
  for (int off = 16; off > 0; off >>= 1) p += __shfl_xor(p, off, 32);
  return p;
#endif
}

// ---------------- sampling helpers ----------------
// jax.image.resize linear 256->512 at integer dest pixel (x,y), edge-clamped
__device__ __forceinline__ float resize256(const float* __restrict__ bb, int x, int y) {
  float sx = 0.5f*(float)x - 0.25f;
  float sy = 0.5f*(float)y - 0.25f;
  float fx = floorf(sx), fy = floorf(sy);
  int x0 = (int)fx, y0 = (int)fy;
  float wx = sx - fx, wy = sy - fy;
  int x0c = x0 < 0 ? 0 : x0;           int x1c = (x0+1) > (ORIG-1) ? (ORIG-1) : (x0+1);
  int y0c = y0 < 0 ? 0 : y0;           int y1c = (y0+1) > (ORIG-1) ? (ORIG-1) : (y0+1);
  float v00 = bb[y0c*ORIG + x0c], v01 = bb[y0c*ORIG + x1c];
  float v10 = bb[y1c*ORIG + x0c], v11 = bb[y1c*ORIG + x1c];
  float a0 = v00 + wx*(v01 - v00);
  float a1 = v10 + wx*(v11 - v10);
  return a0 + wy*(a1 - a0);
}

// bilinear, zeros padding, direct loads from a 512x512 image
__device__ __forceinline__ float bilin_zero512(const float* __restrict__ im, float u, float v) {
  float fu = floorf(u), fv = floorf(v);
  int x0 = (int)fu, y0 = (int)fv;
  float wx = u - fu, wy = v - fv;
  float acc = 0.0f;
  bool yv0 = (unsigned)y0     < (unsigned)UPD;
  bool yv1 = (unsigned)(y0+1) < (unsigned)UPD;
  bool xv0 = (unsigned)x0     < (unsigned)UPD;
  bool xv1 = (unsigned)(x0+1) < (unsigned)UPD;
  if (yv0 && xv0) acc += (1.0f-wx)*(1.0f-wy)*im[y0*UPD + x0];
  if (yv0 && xv1) acc += wx*(1.0f-wy)*im[y0*UPD + x0 + 1];
  if (yv1 && xv0) acc += (1.0f-wx)*wy*im[(y0+1)*UPD + x0];
  if (yv1 && xv1) acc += wx*wy*im[(y0+1)*UPD + x0 + 1];
  return acc;
}

// bilinear, zeros padding, taps come from the on-the-fly resized base image
__device__ __forceinline__ float bilin_zero_resize(const float* __restrict__ bb, float u, float v) {
  float fu = floorf(u), fv = floorf(v);
  int x0 = (int)fu, y0 = (int)fv;
  float wx = u - fu, wy = v - fv;
  float acc = 0.0f;
  bool yv0 = (unsigned)y0     < (unsigned)UPD;
  bool yv1 = (unsigned)(y0+1) < (unsigned)UPD;
  bool xv0 = (unsigned)x0     < (unsigned)UPD;
  bool xv1 = (unsigned)(x0+1) < (unsigned)UPD;
  if (yv0 && xv0) acc += (1.0f-wx)*(1.0f-wy)*resize256(bb, x0,   y0);
  if (yv0 && xv1) acc += wx*(1.0f-wy)       *resize256(bb, x0+1, y0);
  if (yv1 && xv0) acc += (1.0f-wx)*wy       *resize256(bb, x0,   y0+1);
  if (yv1 && xv1) acc += wx*wy              *resize256(bb, x0+1, y0+1);
  return acc;
}

// affine_grid(theta,H,W) + grid_sample pixel mapping: (col w, row h) -> (u,v)
__device__ __forceinline__ void affine_uv(const float* __restrict__ t, float w, float h,
                                          float dim, float* u, float* v) {
  float xs = (2.0f*w + 1.0f)/dim - 1.0f;
  float ys = (2.0f*h + 1.0f)/dim - 1.0f;
  float gx = t[0]*xs + t[1]*ys + t[2];
  float gy = t[3]*xs + t[4]*ys + t[5];
  *u = ((gx + 1.0f)*dim - 1.0f)*0.5f;
  *v = ((gy + 1.0f)*dim - 1.0f)*0.5f;
}

// inner-warped mask value at integer intermediate pixel (ix,iy)
__device__ __forceinline__ float minner_val(const float* __restrict__ mk, int ix, int iy,
                                            const float* __restrict__ t2) {
  float u, v;
  affine_uv(t2, (float)ix, (float)iy, (float)UPD, &u, &v);
  return bilin_zero512(mk, u, v);
}

// double-warped (rotated) mask value at output pixel (w,h), before threshold
__device__ __forceinline__ float mrot_val(const float* __restrict__ mk, int w, int h,
                                          const float* __restrict__ t1,
                                          const float* __restrict__ t2) {
  float u, v;
  affine_uv(t1, (float)w, (float)h, (float)UPD, &u, &v);
  float fu = floorf(u), fv = floorf(v);
  int x0 = (int)fu, y0 = (int)fv;
  float wx = u - fu, wy = v - fv;
  float acc = 0.0f;
  bool yv0 = (unsigned)y0     < (unsigned)UPD;
  bool yv1 = (unsigned)(y0+1) < (unsigned)UPD;
  bool xv0 = (unsigned)x0     < (unsigned)UPD;
  bool xv1 = (unsigned)(x0+1) < (unsigned)UPD;
  if (yv0 && xv0) acc += (1.0f-wx)*(1.0f-wy)*minner_val(mk, x0,   y0,   t2);
  if (yv0 && xv1) acc += wx*(1.0f-wy)       *minner_val(mk, x0+1, y0,   t2);
  if (yv1 && xv0) acc += (1.0f-wx)*wy       *minner_val(mk, x0,   y0+1, t2);
  if (yv1 && xv1) acc += wx*wy              *minner_val(mk, x0+1, y0+1, t2);
  return acc;
}

// ---------------- kernels ----------------
// invert [[a,b,c],[d,e,f],[0,0,1]] -> top 2 rows; zero reduction accumulators
__global__ void k_prep(const float* __restrict__ ss, const float* __restrict__ rot,
                       float* __restrict__ inv1, float* __restrict__ inv2,
                       float* __restrict__ acc1, float* __restrict__ acc2) {
  int t = threadIdx.x;
  if (t < NB) {
    for (int which = 0; which < 2; ++which) {
      const float* th = (which == 0 ? ss : rot) + t*6;
      float* out      = (which == 0 ? inv1 : inv2) + t*6;
      float a = th[0], b = th[1], c = th[2];
      float d = th[3], e = th[4], f = th[5];
      float id = 1.0f / (a*e - b*d);
      out[0] =  e*id;  out[1] = -b*id;  out[2] = (b*f - e*c)*id;
      out[3] = -d*id;  out[4] =  a*id;  out[5] = (d*c - a*f)*id;
    }
  }
  for (int i = t; i < NM*NB*2; i += blockDim.x) acc1[i] = 0.0f;
  for (int i = t; i < NM*NB*3; i += blockDim.x) acc2[i] = 0.0f;
}

// I2 = grid_sample(resize(base), g2), resize fused (256^2 source is L2-resident)
__global__ void k_i2(const float* __restrict__ base, const float* __restrict__ inv2,
                     float* __restrict__ I2) {
  int gid = blockIdx.x * blockDim.x + threadIdx.x;
  int b = gid >> 18;                 // NPIX = 2^18
  int p = gid & (NPIX - 1);
  int h = p >> 9, w = p & (UPD - 1);
  const float* t2 = inv2 + b*6;
  const float* bb = base + (size_t)b * ORIG * ORIG;
  float u, v;
  affine_uv(t2, (float)w, (float)h, (float)UPD, &u, &v);
  I2[gid] = bilin_zero_resize(bb, u, v);
}

// pred = grid_sample(I2, g1); keep an untouched copy + init the carried image
__global__ void k_pred(const float* __restrict__ I2, const float* __restrict__ inv1,
                       float* __restrict__ pred, float* __restrict__ out) {
  int gid = blockIdx.x * blockDim.x + threadIdx.x;
  int b = gid >> 18;
  int p = gid & (NPIX - 1);
  int h = p >> 9, w = p & (UPD - 1);
  const float* t1 = inv1 + b*6;
  const float* im = I2 + (size_t)b * NPIX;
  float u, v;
  affine_uv(t1, (float)w, (float)h, (float)UPD, &u, &v);
  float val = bilin_zero512(im, u, v);
  pred[gid] = val;
  out[gid]  = val;
}

// pass 1: sum(mask), sum(img*mask) for every (m,b); mask recomputed on the fly
__global__ void k_stats1(const float* __restrict__ pred, const float* __restrict__ masks,
                         const float* __restrict__ inv1, const float* __restrict__ inv2,
                         float* __restrict__ acc1) {
  int blk = blockIdx.x;
  int mb = blk >> 4;                 // 16 blocks per (m,b)
  int part = blk & 15;
  int m = mb >> 5, b = mb & (NB - 1);
  const float* t1 = inv1 + b*6;
  const float* t2 = inv2 + b*6;
  const float* mk = masks + (size_t)m * NPIX;
  const float* im = pred + (size_t)b * NPIX;
  float s_mask = 0.0f, s_im = 0.0f;
  int base_p = part * (NPIX / 16);
  for (int i = threadIdx.x; i < NPIX / 16; i += blockDim.x) {
    int p = base_p + i;
    int h = p >> 9, w = p & (UPD - 1);
    float mv = mrot_val(mk, w, h, t1, t2);
    if (mv >= 0.5f) { s_mask += 1.0f; s_im += im[p]; }
  }
  float t0 = wave_sum32(s_mask);
  float t1s = wave_sum32(s_im);
  if ((threadIdx.x & 31) == 0) {
    atomicAdd(&acc1[mb*2 + 0], t0);
    atomicAdd(&acc1[mb*2 + 1], t1s);
  }
}

// pass 2: weighted sums for center of mass (threshold = COEF * masked mean)
__global__ void k_stats2(const float* __restrict__ pred, const float* __restrict__ masks,
                         const float* __restrict__ inv1, const float* __restrict__ inv2,
                         const float* __restrict__ acc1, float* __restrict__ acc2) {
  int blk = blockIdx.x;
  int mb = blk >> 4;
  int part = blk & 15;
  int m = mb >> 5, b = mb & (NB - 1);
  const float* t1 = inv1 + b*6;
  const float* t2 = inv2 + b*6;
  const float* mk = masks + (size_t)m * NPIX;
  const float* im = pred + (size_t)b * NPIX;
  float thresh = COEF * (acc1[mb*2 + 1] / (acc1[mb*2 + 0] + 1e-8f));
  float s_w = 0.0f, s_wr = 0.0f, s_wc = 0.0f;
  int base_p = part * (NPIX / 16);
  for (int i = threadIdx.x; i < NPIX / 16; i += blockDim.x) {
    int p = base_p + i;
    int h = p >> 9, w = p & (UPD - 1);
    float mv = mrot_val(mk, w, h, t1, t2);
    if (mv >= 0.5f) {
      float iv = im[p];
      if (iv > thresh) { s_w += iv; s_wr += iv * (float)h; s_wc += iv * (float)w; }
    }
  }
  float r0 = wave_sum32(s_w);
  float r1 = wave_sum32(s_wr);
  float r2 = wave_sum32(s_wc);
  if ((threadIdx.x & 31) == 0) {
    atomicAdd(&acc2[mb*3 + 0], r0);
    atomicAdd(&acc2[mb*3 + 1], r1);
    atomicAdd(&acc2[mb*3 + 2], r2);
  }
}

// round-half-even COM -> dynamic_slice-clamped crop origins
__global__ void k_final(const float* __restrict__ acc2, int* __restrict__ srsc) {
  int t = threadIdx.x;
  if (t < NM*NB) {
    float s  = acc2[t*3 + 0] + 1e-8f;
    float cx = acc2[t*3 + 1] / s;
    float cy = acc2[t*3 + 2] / s;
    int sr = (int)rintf(cx) - RAD;
    int sc = (int)rintf(cy) - RAD;
    sr = min(max(sr, 0), UPD - CROP);
    sc = min(max(sc, 0), UPD - CROP);
    srsc[t*2 + 0] = sr;
    srsc[t*2 + 1] = sc;
  }
}

// crop -> (disc /= adj) -> affine re-warp -> write back, one workgroup per b.
// The 64x64 f32 tile (row stride 512) is staged into LDS with the Tensor Data
// Mover, then resampled from LDS (removes the in-place RAW hazard).
__global__ void k_crop(float* __restrict__ img, const float* __restrict__ inv1,
                       const int* __restrict__ srsc, const float* __restrict__ adj, int m) {
  __shared__ float tile[CROP*CROP];
  int b = blockIdx.x;
  int mb = m*NB + b;
  int sr = srsc[mb*2 + 0];
  int sc = srsc[mb*2 + 1];
  sr = __builtin_amdgcn_readfirstlane(sr);
  sc = __builtin_amdgcn_readfirstlane(sc);
  float* imgb = img + (size_t)b * NPIX;

#if defined(__gfx1250__) && __has_builtin(__builtin_amdgcn_tensor_load_to_lds)
  if (threadIdx.x < 32) {
    unsigned long long ga = (unsigned long long)(uintptr_t)(imgb + sr*UPD + sc);
    unsigned int lds_off = (unsigned int)(uintptr_t)(&tile[0]);
    v4u g0;
    g0.x = 1u;                                         // count=1, user mode
    g0.y = lds_off;                                    // lds_addr (bytes)
    g0.z = (unsigned int)(ga & 0xffffffffull);         // global_addr[31:0]
    g0.w = (unsigned int)(ga >> 32) | (2u << 30);      // addr[56:32] | type=2
    v8i g1;
    g1[0] = (int)(2u << 16);                           // data_size=4B, no mask
    g1[1] = (int)(((unsigned)UPD & 0xffffu) << 16);    // tensor_dim0 lo16
    g1[2] = (int)(((unsigned)UPD & 0xffffu) << 16);    // dim0 hi=0 | dim1 lo16
    g1[3] = (int)((unsigned)CROP << 16);               // dim1 hi=0 | tile_dim0
    g1[4] = (int)((unsigned)CROP);                     // tile_dim1 | tile_dim2=0
    g1[5] = UPD;                                       // tensor_dim0_stride lo32
    g1[6] = 0;                                         // stride0 hi | stride1 lo
    g1[7] = 0;
    v4i g2 = {0, 0, 0, 0};                             // 2D: higher dims unused
    v4i g3 = {0, 0, 0, 0};
    v8i g4 = {0, 0, 0, 0, 0, 0, 0, 0};                 // clang-23 extra group
    __builtin_amdgcn_tensor_load_to_lds(g0, g1, g2, g3, g4, 0);
    __builtin_amdgcn_s_wait_tensorcnt(0);
  }
  __syncthreads();
#else
  for (int i = threadIdx.x; i < CROP*CROP; i += blockDim.x) {
    int ly = i >> 6, lx = i & 63;
    tile[i] = imgb[(sr + ly)*UPD + (sc + lx)];
  }
  __syncthreads();
#endif

  float inva = 1.0f / adj[b];
  const float* t1 = inv1 + b*6;
  float T0 = t1[0], T1 = t1[1], T2 = t1[2], T3 = t1[3], T4 = t1[4], T5 = t1[5];

  for (int i = threadIdx.x; i < CROP*CROP; i += blockDim.x) {
    int ly = i >> 6, lx = i & 63;
    float xs = (2.0f*lx + 1.0f)*(1.0f/CROP) - 1.0f;
    float ys = (2.0f*ly + 1.0f)*(1.0f/CROP) - 1.0f;
    float gx = T0*xs + T1*ys + T2;
    float gy = T3*xs + T4*ys + T5;
    float u = ((gx + 1.0f)*CROP - 1.0f)*0.5f;
    float v = ((gy + 1.0f)*CROP - 1.0f)*0.5f;
    float fu = floorf(u), fv = floorf(v);
    int x0 = (int)fu, y0 = (int)fv;
    float wx = u - fu, wy = v - fv;
    float acc = 0.0f;
    #pragma unroll
    for (int dy = 0; dy < 2; ++dy) {
      #pragma unroll
      for (int dx = 0; dx < 2; ++dx) {
        int ix = x0 + dx, iy = y0 + dy;
        if ((unsigned)ix < (unsigned)CROP && (unsigned)iy < (unsigned)CROP) {
          float val = tile[iy*CROP + ix];
          int ddy = iy - RAD, ddx = ix - RAD;
          if (ddy*ddy + ddx*ddx <= DOT2) val *= inva;  // SQ_MASK divide on source
          float wgt = (dx ? wx : 1.0f - wx) * (dy ? wy : 1.0f - wy);
          acc += wgt * val;
        }
      }
    }
    imgb[(sr + ly)*UPD + (sc + lx)] = acc;
  }
}

// ---------------- host launch ----------------
extern "C" void kernel_launch(void* const* d_in, const int* in_sizes, int n_in,
                              void* d_out, int out_size, void* d_ws, size_t ws_size,
                              hipStream_t stream) {
  const float* base  = (const float*)d_in[0];   // [32,1,256,256]
  const float* ss    = (const float*)d_in[1];   // [32,2,3]
  const float* rot   = (const float*)d_in[2];   // [32,2,3]
  const float* adj   = (const float*)d_in[3];   // [32]
  const float* masks = (const float*)d_in[4];   // [4,512,512]
  float* out = (float*)d_out;                   // [32,1,512,512]

  float* pred = (float*)d_ws;                     // 32 MB original pred_input
  float* I2   = pred + (size_t)NB * NPIX;         // 32 MB inner-warp intermediate
  float* inv1 = I2   + (size_t)NB * NPIX;
  float* inv2 = inv1 + NB*6;
  float* acc1 = inv2 + NB*6;                      // [M*B][2]
  float* acc2 = acc1 + NM*NB*2;                   // [M*B][3]
  int*   srsc = (int*)(acc2 + NM*NB*3);           // [M*B][2]

  k_prep<<<1, 256, 0, stream>>>(ss, rot, inv1, inv2, acc1, acc2);

  int blocks = (NB * NPIX) / 256;                 // 32768
  k_i2  <<<blocks, 256, 0, stream>>>(base, inv2, I2);
  k_pred<<<blocks, 256, 0, stream>>>(I2, inv1, pred, out);

  k_stats1<<<NM*NB*16, 256, 0, stream>>>(pred, masks, inv1, inv2, acc1);
  k_stats2<<<NM*NB*16, 256, 0, stream>>>(pred, masks, inv1, inv2, acc1, acc2);
  k_final <<<1, 128, 0, stream>>>(acc2, srsc);

  for (int m = 0; m < NM; ++m)                    // lax.scan over masks: sequential
    k_crop<<<NB, 256, 0, stream>>>(out, inv1, srsc, adj, m);
}